// Net_56650618635136
// MI455X (gfx1250) — compile-verified
//
#include <hip/hip_runtime.h>

typedef __attribute__((ext_vector_type(2))) float v2f;
typedef __attribute__((ext_vector_type(8))) float v8f;

#define IN_CH 128

// One wave32 computes p[] and c[] for a tile of 16 nodes with 32 chained
// V_WMMA_F32_16X16X4_F32 ops over K=128.
//   A (16x4 f32)  = weights: row 0 = Wp, row 1 = Wc, rows 2..15 = 0
//                   (read from LDS: [Wp | pad | Wc | pad | zeros], branch-free)
//   B (4x16 f32)  = x tile:  column n = node base+n
//   C/D (16x16)   : D[0,n] = p[base+n], D[1,n] = c[base+n]; bias preloaded.
// Result: lanes 0..15 hold p in acc[0], c in acc[1] -> coalesced stores.
__global__ __launch_bounds__(256) void node_proj_wmma(
    const float* __restrict__ x,
    const float* __restrict__ Wp, const float* __restrict__ bp,
    const float* __restrict__ Wc, const float* __restrict__ bc,
    float* __restrict__ outp, float* __restrict__ outc, int n_nodes)
{
  // LDS: Wp at [0..127], Wc at [132..259], zeros at [264..391] (pad vs banks)
  __shared__ float wlds[392];
  const int tid = (int)threadIdx.x;
  if (tid < 128) {
    wlds[tid]       = Wp[tid];
    wlds[132 + tid] = Wc[tid];
    wlds[264 + tid] = 0.0f;
  }
  __syncthreads();

  const int lane = tid & 31;
  const int wave = (int)(blockIdx.x * (blockDim.x >> 5)) + (tid >> 5);
  const int base = wave << 4;                 // 16 nodes per wave
  if (base >= n_nodes) return;                // wave-uniform exit (EXEC stays full)

  const int col   = lane & 15;                // node-in-tile (B column)
  const int khalf = lane >> 4;                // K sub-pair select

  int row = base + col;
  if (row >= n_nodes) row = n_nodes - 1;      // clamp loads for ragged tail
  const float* xrow = x + (size_t)row * IN_CH + 2 * khalf;

  const int woff = (col == 0) ? 0 : ((col == 1) ? 132 : 264);
  const float* wrow = &wlds[woff + 2 * khalf];

  const float bpv = bp[0];
  const float bcv = bc[0];
  v8f acc;
  acc[0] = (khalf == 0) ? bpv : 0.0f;         // C[0,n] = bp
  acc[1] = (khalf == 0) ? bcv : 0.0f;         // C[1,n] = bc
  #pragma unroll
  for (int r = 2; r < 8; ++r) acc[r] = 0.0f;

  #pragma unroll
  for (int t = 0; t < IN_CH / 4; ++t) {
    const int k0 = 4 * t;
    v2f a = *(const v2f*)(wrow + k0);         // ds_load_b64 (branch-free)
    v2f b = *(const v2f*)(xrow + k0);         // global_load_b64, 8B aligned
    // 8 args: (neg_a, A, neg_b, B, c_mod, C, reuse_a, reuse_b)
    acc = __builtin_amdgcn_wmma_f32_16x16x4_f32(
        false, a, false, b, (short)0, acc, false, false);
  }

  // D[0,n] / D[1,n] live in acc[0]/acc[1] of lanes 0..15.
  if (khalf == 0) {
    const int n = base + col;
    if (n < n_nodes) {
      outp[n] = acc[0];
      outc[n] = acc[1];
    }
  }
}

// out[e] = |p[dst[e]] - c[src[e]]| * W1[0] + b1[0]
// adjs is int64 [2, E]: row 0 = src, row 1 = dst. p/c (800 KB) live in L2.
__global__ __launch_bounds__(256) void edge_out_kernel(
    const long long* __restrict__ adjs,
    const float* __restrict__ p, const float* __restrict__ c,
    const float* __restrict__ W1, const float* __restrict__ b1,
    float* __restrict__ out, int n_edges)
{
  const float w1 = W1[0];
  const float bb = b1[0];
  int e = (int)(blockIdx.x * blockDim.x + threadIdx.x);
  const int stride = (int)(gridDim.x * blockDim.x);
  for (; e < n_edges; e += stride) {
    const long long s = adjs[e];                    // src
    const long long d = adjs[(size_t)n_edges + e];  // dst
    const float v = __builtin_fabsf(p[d] - c[s]);
    out[e] = v * w1 + bb;
  }
}

extern "C" void kernel_launch(void* const* d_in, const int* in_sizes, int n_in,
                              void* d_out, int out_size, void* d_ws, size_t ws_size,
                              hipStream_t stream) {
  const float*     x    = (const float*)d_in[0];
  const long long* adjs = (const long long*)d_in[1];
  const float*     Wp   = (const float*)d_in[2];
  const float*     bp   = (const float*)d_in[3];
  const float*     Wc   = (const float*)d_in[4];
  const float*     bc   = (const float*)d_in[5];
  const float*     W1   = (const float*)d_in[6];
  const float*     b1   = (const float*)d_in[7];
  float*           out  = (float*)d_out;

  const int n_nodes = in_sizes[0] / IN_CH;    // 100000
  const int n_edges = in_sizes[1] / 2;        // 600000

  float* p = (float*)d_ws;                    // [n_nodes]
  float* c = p + n_nodes;                     // [n_nodes]

  // Node projection: 16 nodes per wave, 8 waves per 256-thread block.
  const int tiles   = (n_nodes + 15) / 16;
  const int wpb     = 256 / 32;
  const int blocksA = (tiles + wpb - 1) / wpb;
  node_proj_wmma<<<blocksA, 256, 0, stream>>>(x, Wp, bp, Wc, bc, p, c, n_nodes);

  // Edge gather + affine.
  int blocksB = (n_edges + 255) / 256;
  if (blocksB > 16384) blocksB = 16384;
  edge_out_kernel<<<blocksB, 256, 0, stream>>>(adjs, p, c, W1, b1, out, n_edges);
}